// DiffusionModel_27479200759936
// MI455X (gfx1250) — compile-verified
//
#include <hip/hip_runtime.h>
#include <hip/hip_bf16.h>

typedef __attribute__((ext_vector_type(2))) float v2f;
typedef __attribute__((ext_vector_type(4))) float v4f;
typedef __attribute__((ext_vector_type(8))) float v8f;

#define NCELLS 4096
#define MCHEM  8

// One wave (32 lanes) computes a 16-row (i) tile of the output, looping over
// all j in 16-column tiles. V_WMMA_F32_16X16X4_F32 produces the full 16x16
// tile of (dist^2 + eps) in one instruction:
//   D = A(-2*p_i, 16x4) x B(p_j, 4x16) + C(|p_i|^2 + |p_j|^2 + eps)
__global__ __launch_bounds__(256, 1) void diffusion_field_kernel(
    const float* __restrict__ pos,        // [N,3]
    const float* __restrict__ radius,     // [N]
    const float* __restrict__ secretion,  // [N,M]
    const float* __restrict__ diff_coef,  // [M]
    const float* __restrict__ degrade,    // [M]
    const unsigned char* __restrict__ active, // [N] (jnp bool, 1 byte)
    float* __restrict__ out)              // [N,M]
{
    const int lane       = threadIdx.x & 31;
    const int waveGlobal = blockIdx.x * (blockDim.x >> 5) + (threadIdx.x >> 5);
    const int iTile      = waveGlobal * 16;          // base i row of this wave
    const int jl         = lane & 15;                // j column owned by lane
    const bool hiHalf    = lane >= 16;

    // Per-chemical constants (loop invariant; tiny, L2-resident).
    // exp(-d/lam_m) = exp2(d * klam_m), klam_m = -log2(e)*sqrt(k_m/D_m)
    // 1/(4*pi*D_m*d) = coef_m * (1/d)
    float coef[MCHEM], klam[MCHEM];
#pragma unroll
    for (int m = 0; m < MCHEM; ++m) {
        float D = diff_coef[m];
        float K = degrade[m];
        coef[m] = 1.0f / (12.566370614359172f * D);
        klam[m] = -1.4426950408889634f * __builtin_amdgcn_sqrtf(K / D);
    }

    // ---- A matrix: -2 * p_i, ISA 32-bit 16x4 A layout ----
    // lanes 0-15 : VGPR0 = A[M=lane, K=0], VGPR1 = A[M=lane, K=1]
    // lanes 16-31: VGPR0 = A[M=lane-16, K=2], VGPR1 = A[M=lane-16, K=3] (=0)
    {
        // fallthrough into Amat init below
    }
    const int iA = iTile + jl;
    const float ax = pos[3 * iA + 0];
    const float ay = pos[3 * iA + 1];
    const float az = pos[3 * iA + 2];
    v2f Amat;
    Amat.x = hiHalf ? (-2.0f * az) : (-2.0f * ax);
    Amat.y = hiHalf ? 0.0f         : (-2.0f * ay);

    // ---- cA[r] = |p_i|^2 + eps for this lane's 8 D-row slots ----
    // D element (VGPR r, lane) <-> M = r + (hiHalf?8:0), N = jl
    const int iRowBase = iTile + (hiHalf ? 8 : 0);
    float cA[8];
#pragma unroll
    for (int r = 0; r < 8; ++r) {
        const int i = iRowBase + r;
        const float x = pos[3 * i + 0];
        const float y = pos[3 * i + 1];
        const float z = pos[3 * i + 2];
        cA[r] = fmaf(x, x, fmaf(y, y, z * z)) + 1e-8f;
    }

    float acc[8][MCHEM];
#pragma unroll
    for (int r = 0; r < 8; ++r)
#pragma unroll
        for (int m = 0; m < MCHEM; ++m) acc[r][m] = 0.0f;

    for (int jBase = 0; jBase < NCELLS; jBase += 16) {
        const int j = jBase + jl;

        // prefetch next tile's positions (speculative; dropped past the end)
        __builtin_prefetch(pos + 3 * (j + 16), 0, 0);

        const float bx = pos[3 * j + 0];
        const float by = pos[3 * j + 1];
        const float bz = pos[3 * j + 2];
        const float pj2 = fmaf(bx, bx, fmaf(by, by, bz * bz));
        const float rad = radius[j];
        const float act = active[j] ? 1.0f : 0.0f;

        // per-j source with 1/(4*pi*D_m) and activity folded in
        const float* srow = secretion + j * MCHEM;
        float sp[MCHEM];
#pragma unroll
        for (int m = 0; m < MCHEM; ++m) sp[m] = srow[m] * act * coef[m];

        // ---- B matrix: p_j, ISA 32-bit 4x16 B layout (mirror of A) ----
        v2f Bmat;
        Bmat.x = hiHalf ? bz : bx;
        Bmat.y = hiHalf ? 0.0f : by;

        // ---- C = |p_i|^2 + |p_j|^2 + eps ----
        v8f Cm;
#pragma unroll
        for (int r = 0; r < 8; ++r) Cm[r] = cA[r] + pj2;

        // D[r] = dist^2 + eps for (i = iRowBase + r, j)
        v8f Dv = __builtin_amdgcn_wmma_f32_16x16x4_f32(
            /*neg_a=*/false, Amat, /*neg_b=*/false, Bmat,
            /*c_mod=*/(short)0, Cm, /*reuse_a=*/false, /*reuse_b=*/false);

#pragma unroll
        for (int r = 0; r < 8; ++r) {
            float dist = __builtin_amdgcn_sqrtf(Dv[r]);
            dist = fmaxf(dist, rad);
            const float invd = __builtin_amdgcn_rcpf(dist);
#pragma unroll
            for (int m = 0; m < MCHEM; ++m) {
                const float e = __builtin_amdgcn_exp2f(dist * klam[m]);
                acc[r][m] = fmaf(e * invd, sp[m], acc[r][m]);
            }
        }
    }

    // Reduce over the 16 j-columns: xor-shuffles with masks 1,2,4,8 stay
    // inside each 16-lane half of the wave32.
#pragma unroll
    for (int r = 0; r < 8; ++r) {
#pragma unroll
        for (int m = 0; m < MCHEM; ++m) {
            float v = acc[r][m];
            v += __shfl_xor(v, 1, 32);
            v += __shfl_xor(v, 2, 32);
            v += __shfl_xor(v, 4, 32);
            v += __shfl_xor(v, 8, 32);
            acc[r][m] = v;
        }
    }

    // Lane 0 writes rows iTile+0..7, lane 16 writes rows iTile+8..15.
    if (jl == 0) {
#pragma unroll
        for (int r = 0; r < 8; ++r) {
            float* o = out + (size_t)(iRowBase + r) * MCHEM;
            v4f lo = { acc[r][0], acc[r][1], acc[r][2], acc[r][3] };
            v4f hi = { acc[r][4], acc[r][5], acc[r][6], acc[r][7] };
            *(v4f*)(o + 0) = lo;
            *(v4f*)(o + 4) = hi;
        }
    }
}

extern "C" void kernel_launch(void* const* d_in, const int* in_sizes, int n_in,
                              void* d_out, int out_size, void* d_ws, size_t ws_size,
                              hipStream_t stream) {
    const float* pos       = (const float*)d_in[0];   // position [N,3]
    const float* radius    = (const float*)d_in[1];   // radius [N]
    const float* secretion = (const float*)d_in[2];   // secretion [N,M]
    const float* Dm        = (const float*)d_in[3];   // diffusion_coefs [M]
    const float* Km        = (const float*)d_in[4];   // degradation_rates [M]
    const unsigned char* active = (const unsigned char*)d_in[5]; // bool [N]
    float* out = (float*)d_out;

    // 4096 rows / 16 per wave / 8 waves per block = 32 blocks of 256 threads
    dim3 block(256);
    dim3 grid(NCELLS / (16 * 8));
    hipLaunchKernelGGL(diffusion_field_kernel, grid, block, 0, stream,
                       pos, radius, secretion, Dm, Km, active, out);
}